// LabelSmoothLoss_283467841546
// MI455X (gfx1250) — compile-verified
//
#include <hip/hip_runtime.h>
#include <math.h>

// ---------------------------------------------------------------------------
// LabelSmoothLoss for MI455X (gfx1250, wave32).
//
// candidates = predicts @ (mask*adj + diag fixup) / counts dominates:
// stream adj (604MB f32) + adj_mask (151MB u8) once -> ~32us floor at
// 23.3 TB/s. MACs go to V_WMMA_F32_16X16X4_F32 (full f32 precision) so the
// VALU budget per streamed byte stays under the HBM issue-rate requirement.
// ---------------------------------------------------------------------------

typedef float v2f __attribute__((ext_vector_type(2)));
typedef float v8f __attribute__((ext_vector_type(8)));

constexpr int LDIM  = 12288;  // labels
constexpr int NB    = 8;      // batch
constexpr int NCOL  = 16;     // columns per workgroup (WMMA N)
constexpr int NWAVE = 8;      // waves per workgroup
constexpr int KSTEP = NWAVE * 4;  // K rows consumed per loop iter (all waves)

// ---------------------------------------------------------------------------
// Prologue: zero-padded A operand. Ppad[16][LDIM]: rows 0-7 = predicts,
// rows 8-15 = 0. Removes the per-iteration v_pk_mul_f32 A-fixup in stage 1.
// ---------------------------------------------------------------------------
__global__ __launch_bounds__(256) void ls_pad(const float* __restrict__ P,
                                              float* __restrict__ Ppad) {
  const int idx = blockIdx.x * 256 + threadIdx.x;  // 16*LDIM threads
  Ppad[idx] = (idx < NB * LDIM) ? P[idx] : 0.0f;
}

// ---------------------------------------------------------------------------
// Stage 1: candidates[b][j] = (sum_li m*adj[li][j]*P[b][li] + (1-diag)*P[b][j])
//                             / (sum_li m[li][j] + (1-diag))
// One WG per 16-column tile; 8 waves split the K range. Uniform, compile-time
// trip count (384) so the loop unrolls and uses scalar branch control.
// ---------------------------------------------------------------------------
__global__ __launch_bounds__(256, 2) void ls_stage1(
    const float* __restrict__ Ppad,         // [16][LDIM] zero-padded
    const float* __restrict__ P,            // [NB][LDIM] original
    const unsigned char* __restrict__ msk,  // [LDIM][LDIM] (jax bool = 1 byte)
    const float* __restrict__ adj,          // [LDIM][LDIM]
    float* __restrict__ cand)               // [NB][LDIM] (workspace)
{
  const int tid  = threadIdx.x;
  const int lane = tid & 31;
  const int wave = tid >> 5;
  const int col  = lane & 15;   // N within tile; also A-matrix row M
  const int half = lane >> 4;   // 0: K={0,1}, 1: K={2,3}  (ISA f32 A/B layout)
  const int c0   = blockIdx.x * NCOL;
  const int j    = c0 + col;    // absolute column

  __shared__ float lds_c[NWAVE][16][16];  // per-wave C tiles
  __shared__ float lds_cnt[NWAVE][32];    // per-lane mask counts
  __shared__ float lds_fin[2][16];        // [0]=counts, [1]=diag

  // Per-lane streaming pointers (K start = wave*4 + half*2).
  const int r0 = wave * 4 + half * 2;
  const unsigned char* mp = msk + (size_t)r0 * LDIM + j;
  const float*         ap = adj + (size_t)r0 * LDIM + j;
  const float*         pp = Ppad + (size_t)col * LDIM + r0;  // A row M=col

  v8f   acc = {0.f, 0.f, 0.f, 0.f, 0.f, 0.f, 0.f, 0.f};
  float cnt = 0.0f;

#pragma unroll 4
  for (int it = 0; it < LDIM / KSTEP; ++it) {
    // B fragment: b_eff rows r0,r0+1 at column j (streamed once -> NT hint)
    const float m0 = (float)__builtin_nontemporal_load(mp);
    const float m1 = (float)__builtin_nontemporal_load(mp + LDIM);
    const float a0 = __builtin_nontemporal_load(ap);
    const float a1 = __builtin_nontemporal_load(ap + LDIM);
    v2f bfrag;
    bfrag.x = m0 * a0;
    bfrag.y = m1 * a1;
    cnt += m0 + m1;  // counts partial

    // A fragment: Ppad[M=col][K=r0..r0+1] -> one global_load_b64, L2-hot
    v2f afrag;
    afrag.x = pp[0];
    afrag.y = pp[1];

    // D(16x16,f32) += A(16x4,f32) x B(4x16,f32)
    acc = __builtin_amdgcn_wmma_f32_16x16x4_f32(
        /*neg_a=*/false, afrag, /*neg_b=*/false, bfrag,
        /*c_mod=*/(short)0, acc, /*reuse_a=*/false, /*reuse_b=*/false);

    mp += (size_t)KSTEP * LDIM;
    ap += (size_t)KSTEP * LDIM;
    pp += KSTEP;
  }

  // C/D layout: VGPR v -> M=v (lanes 0-15), M=v+8 (lanes 16-31)
#pragma unroll
  for (int v = 0; v < 8; ++v)
    lds_c[wave][half * 8 + v][col] = acc[v];
  lds_cnt[wave][lane] = cnt;
  __syncthreads();

  // counts + diagonal fixup per column (16 threads)
  if (tid < 16) {
    float s = 0.0f;
#pragma unroll
    for (int w = 0; w < NWAVE; ++w)
      s += lds_cnt[w][tid] + lds_cnt[w][tid + 16];
    const int   jj   = c0 + tid;
    const float diag = (float)msk[(size_t)jj * LDIM + jj];
    lds_fin[0][tid] = s + (1.0f - diag);  // counts (always >= 1)
    lds_fin[1][tid] = diag;
  }
  __syncthreads();

  // reduce 8 wave-tiles, add identity term, divide, store (128 threads)
  if (tid < 128) {
    const int b = tid >> 4, c = tid & 15;
    float s = 0.0f;
#pragma unroll
    for (int w = 0; w < NWAVE; ++w) s += lds_c[w][b][c];
    const int jj = c0 + c;
    s += (1.0f - lds_fin[1][c]) * P[b * LDIM + jj];
    cand[(size_t)b * LDIM + jj] = s / lds_fin[0][c];
  }
}

// ---------------------------------------------------------------------------
// Stage 2: per-block partial sums of diff^2 per batch row (deterministic
// fixed-order tree reduction; no float atomics so graph replays are bitwise
// stable). diff[b][j] = P[b][j] - sum_k S[b][k]*cand[k][j]
// ---------------------------------------------------------------------------
__global__ __launch_bounds__(256) void ls_stage2(
    const float* __restrict__ P, const float* __restrict__ S,
    const float* __restrict__ cand, float* __restrict__ partial)
{
  const int tid = threadIdx.x;
  const int j   = blockIdx.x * 256 + tid;

  __shared__ float red[NB][256];  // [b][tid], conflict-free across lanes

  float c[NB];
#pragma unroll
  for (int k = 0; k < NB; ++k) c[k] = cand[(size_t)k * LDIM + j];

#pragma unroll
  for (int b = 0; b < NB; ++b) {
    float d = P[(size_t)b * LDIM + j];
#pragma unroll
    for (int k = 0; k < NB; ++k) d -= S[b * NB + k] * c[k];
    red[b][tid] = d * d;
  }
  __syncthreads();

  for (int s = 128; s > 0; s >>= 1) {
    if (tid < s) {
#pragma unroll
      for (int b = 0; b < NB; ++b) red[b][tid] += red[b][tid + s];
    }
    __syncthreads();
  }
  if (tid < NB) partial[blockIdx.x * NB + tid] = red[tid][0];
}

// ---------------------------------------------------------------------------
// Stage 3: final 48x8 reduce + sqrt + validity-masked mean (one thread,
// fixed order -> deterministic).
// ---------------------------------------------------------------------------
__global__ void ls_stage3(const float* __restrict__ S,
                          const float* __restrict__ partial, int nwg,
                          float* __restrict__ out)
{
  if (threadIdx.x == 0 && blockIdx.x == 0) {
    float total = 0.0f, cntv = 0.0f;
    for (int b = 0; b < NB; ++b) {
      float sum = 0.0f;
      for (int w = 0; w < nwg; ++w) sum += partial[w * NB + b];
      float rs = 0.0f;
      for (int k = 0; k < NB; ++k) rs += S[b * NB + k];
      if (rs != 0.0f) {
        total += sqrtf(sum);
        cntv += 1.0f;
      }
    }
    out[0] = (cntv == 0.0f) ? 0.0f : total / fmaxf(cntv, 1.0f);
  }
}

// ---------------------------------------------------------------------------
extern "C" void kernel_launch(void* const* d_in, const int* in_sizes, int n_in,
                              void* d_out, int out_size, void* d_ws,
                              size_t ws_size, hipStream_t stream) {
  const float*         P   = (const float*)d_in[0];          // [8][12288]
  const float*         S   = (const float*)d_in[1];          // [8][8]
  const float*         adj = (const float*)d_in[2];          // [L][L]
  const unsigned char* msk = (const unsigned char*)d_in[3];  // [L][L] bool
  float*               out = (float*)d_out;

  float* cand    = (float*)d_ws;                  // 8*LDIM floats
  float* partial = cand + (size_t)NB * LDIM;      // 48*8 floats
  float* Ppad    = partial + (size_t)(LDIM / 256) * NB;  // 16*LDIM floats

  ls_pad<<<(16 * LDIM) / 256, 256, 0, stream>>>(P, Ppad);
  ls_stage1<<<LDIM / NCOL, 256, 0, stream>>>(Ppad, P, msk, adj, cand);
  ls_stage2<<<LDIM / 256, 256, 0, stream>>>(P, S, cand, partial);
  ls_stage3<<<1, 32, 0, stream>>>(S, partial, LDIM / 256, out);
}